// IntertwiningTransformer_68779606278684
// MI455X (gfx1250) — compile-verified
//
#include <hip/hip_runtime.h>
#include <math.h>

typedef __attribute__((ext_vector_type(2))) float v2f;
typedef __attribute__((ext_vector_type(4))) float v4f;
typedef __attribute__((ext_vector_type(8))) float v8f;
typedef __attribute__((ext_vector_type(4))) unsigned int u32x4;
typedef __attribute__((ext_vector_type(8))) unsigned int u32x8;

#define BDIM 256
#define NB 256   // batches
#define NA 512   // atoms
#define AD 64    // atom_dim
#define NH 256   // hidden

// ---- LDS layout (floats), dynamic shared memory ----
constexpr int SE_STRIDE  = 68;                  // 64 + 4 pad (TDM pad_amount=4dw/64dw)
constexpr int SG_STRIDE  = 76;                  // 64 + 12 pad: halves land >=24 banks apart
constexpr int SE_ELEMS   = NA * SE_STRIDE;      // 34816 floats (136 KB) for E=cpd[b]
constexpr int OFF_G      = SE_ELEMS;            // [64*76]  G = W W^T (padded)
constexpr int OFF_U      = OFF_G + AD * SG_STRIDE; // [68] u2[0..63] = 2*W b^T, [64] = |b|^2
constexpr int OFF_SSQ    = OFF_U + 68;          // [512] per-atom |x|^2
constexpr int OFF_ATT    = OFF_SSQ + NA;        // [512] exp(norm - max)
constexpr int OFF_RED    = OFF_ATT + NA;        // [32]  block reduction scratch
constexpr int OFF_C      = OFF_RED + 32;        // [256] 4 partial chunks x 64
constexpr int OFF_C2     = OFF_C + 256;         // [64]  weighted cpd vector
constexpr int OFF_POOL   = OFF_C2 + 64;         // [256] pooled hidden vector
constexpr int OFF_H      = OFF_POOL + NH;       // [8]   fc1 output
constexpr int LDS_FLOATS = OFF_H + 8;           // ~166 KB total (within 320 KB WGP)

// ws layout (floats): [0,4096) = G (64x64, row-major), [4096,4160) = 2*u, [4160] = |b|^2
constexpr int WS_G = 0;
constexpr int WS_U = 4096;
constexpr int WS_C0 = 4160;
constexpr int WS_FLOATS = 4161;

// ============ prep kernel: G = W W^T, u2 = 2 W b^T, c0 = |b|^2 ============
__global__ __launch_bounds__(BDIM)
void it_prep_kernel(const float* __restrict__ Wl, const float* __restrict__ blv,
                    float* __restrict__ ws) {
    const int bi = blockIdx.x;
    const int t  = threadIdx.x;
    if (bi < 16) {                        // 16 blocks x 256 threads = 4096 G entries
        int flat = bi * 256 + t;
        int i = flat >> 6, j = flat & 63;
        const float* wi = Wl + i * NH;
        const float* wj = Wl + j * NH;
        float acc = 0.f;
        for (int n = 0; n < NH; ++n) acc = fmaf(wi[n], wj[n], acc);
        ws[WS_G + flat] = acc;
    } else {
        if (t < 64) {
            const float* wj = Wl + t * NH;
            float acc = 0.f;
            for (int n = 0; n < NH; ++n) acc = fmaf(wj[n], blv[n], acc);
            ws[WS_U + t] = 2.0f * acc;
        } else if (t == 64) {
            float acc = 0.f;
            for (int n = 0; n < NH; ++n) acc = fmaf(blv[n], blv[n], acc);
            ws[WS_C0] = acc;
        }
    }
}

// ============ main fused kernel: one block per batch ============
__global__ __launch_bounds__(BDIM)
void it_fused_kernel(const float* __restrict__ cpd,   // [B, A, 64]
                     const float* __restrict__ Wl,    // [64, 256]
                     const float* __restrict__ blv,   // [256]
                     const float* __restrict__ W1,    // [256, 8]
                     const float* __restrict__ b1v,   // [8]
                     const float* __restrict__ W2,    // [8, 2]
                     const float* __restrict__ b2v,   // [2]
                     const float* __restrict__ ws,    // G / u2 / c0
                     float* __restrict__ out)         // [B, 2]
{
    extern __shared__ float smem[];
    float* sE = smem;

    const int tid  = threadIdx.x;
    const int lane = tid & 31;
    const int wave = tid >> 5;
    const int l16  = lane & 15;
    const int hsel = lane >> 4;
    const int b    = blockIdx.x;

    const float* cb = cpd + (size_t)b * NA * AD;

    // ---------- TDM: DMA whole E tile (512x64 f32) global->LDS, hw-padded rows ----------
    // D# per CDNA5 ISA: 2D tensor -> groups 0+1 only.
    if (wave == 0) {
        unsigned long long ga = (unsigned long long)(uintptr_t)cb;
        u32x4 g0;
        g0[0] = 1u;                                        // count=1, user desc
        g0[1] = (unsigned)(uintptr_t)sE;                   // lds_addr (AS3 offset)
        g0[2] = (unsigned)(ga & 0xffffffffu);              // global_addr[31:0]
        g0[3] = (unsigned)((ga >> 32) & 0x01ffffffu)       // global_addr[56:32]
              | (2u << 30);                                // type=2 ("image")
        u32x8 g1;
        g1[0] = (2u << 16)      // data_size = 4 bytes
              | (1u << 20)      // pad_enable
              | (5u << 22)      // pad_interval: every 64 dwords (one row)
              | (3u << 25);     // pad_amount: 4 dwords -> row stride 68 floats
        g1[1] = (unsigned)(AD) << 16;     // tensor_dim0 = 64
        g1[2] = (unsigned)(NA) << 16;     // tensor_dim0 hi=0 | tensor_dim1 = 512
        g1[3] = (unsigned)(AD) << 16;     // tensor_dim1 hi=0 | tile_dim0 = 64
        g1[4] = (unsigned)(NA);           // tile_dim1 = 512 | tile_dim2 = 0
        g1[5] = (unsigned)(AD);           // tensor_dim0_stride = 64 (lo)
        g1[6] = 0u;                       // stride hi | tensor_dim1_stride lo
        g1[7] = 0u;
        asm volatile("tensor_load_to_lds %0, %1" :: "s"(g0), "s"(g1) : "memory");
    }

    // ---------- meanwhile: stage G (padded) and u2/c0 into LDS ----------
    {
        const v4f* g = (const v4f*)(ws + WS_G);            // 1024 float4, 4 per thread
        #pragma unroll
        for (int i = 0; i < 4; ++i) {
            int j4 = tid + BDIM * i;
            int r  = j4 >> 4;                              // 16 float4 per row
            int c4 = j4 & 15;
            *(v4f*)&smem[OFF_G + r * SG_STRIDE + c4 * 4] = g[j4];
        }
        if (tid < 65) smem[OFF_U + tid] = ws[WS_U + tid];  // u2[64] + c0
    }
    if (wave == 0) __builtin_amdgcn_s_wait_tensorcnt(0);
    __syncthreads();

    // ---------- phase 1: ssq[a] = e_a G e_a^T + 2 e_a.u + |b|^2 via WMMA ----------
    const float c0 = smem[OFF_U + 64];
    for (int t4 = 0; t4 < 4; ++t4) {
        const int m0 = (wave * 4 + t4) * 16;
        // A fragments: lane holds (row = m0+l16, K = 4kk+2*hsel, +1)
        v2f a[16];
        #pragma unroll
        for (int kk = 0; kk < 16; ++kk)
            a[kk] = *(const v2f*)&sE[(m0 + l16) * SE_STRIDE + 4 * kk + 2 * hsel];

        float dacc[8] = {0.f, 0.f, 0.f, 0.f, 0.f, 0.f, 0.f, 0.f};
        #pragma unroll
        for (int n0 = 0; n0 < AD; n0 += 16) {
            v8f c = {0.f, 0.f, 0.f, 0.f, 0.f, 0.f, 0.f, 0.f};
            #pragma unroll
            for (int kk = 0; kk < 16; ++kk) {
                int r0 = 4 * kk + 2 * hsel;
                v2f bb;
                bb.x = smem[OFF_G + r0 * SG_STRIDE + n0 + l16];
                bb.y = smem[OFF_G + (r0 + 1) * SG_STRIDE + n0 + l16];
                c = __builtin_amdgcn_wmma_f32_16x16x4_f32(
                        false, a[kk], false, bb, (short)0, c, false, false);
            }
            // row-dot with E, folding the linear term via (X + 2u_n) * e
            float u2n = smem[OFF_U + n0 + l16];
            #pragma unroll
            for (int r = 0; r < 8; ++r) {
                float e = sE[(m0 + r + 8 * hsel) * SE_STRIDE + n0 + l16];
                dacc[r] = fmaf(c[r] + u2n, e, dacc[r]);
            }
        }
        #pragma unroll
        for (int r = 0; r < 8; ++r) {
            float v = dacc[r];
            v += __shfl_xor(v, 1, 32);
            v += __shfl_xor(v, 2, 32);
            v += __shfl_xor(v, 4, 32);
            v += __shfl_xor(v, 8, 32);
            if (l16 == 0) smem[OFF_SSQ + m0 + r + 8 * hsel] = v + c0;
        }
    }
    __syncthreads();

    // ---------- phase 2: softmax over atom L2 norms ----------
    float nv0 = sqrtf(smem[OFF_SSQ + tid]);
    float nv1 = sqrtf(smem[OFF_SSQ + tid + 256]);
    float m = fmaxf(nv0, nv1);
    #pragma unroll
    for (int sh = 16; sh > 0; sh >>= 1) m = fmaxf(m, __shfl_xor(m, sh, 32));
    if (lane == 0) smem[OFF_RED + wave] = m;
    __syncthreads();
    float gmax = smem[OFF_RED];
    #pragma unroll
    for (int i = 1; i < 8; ++i) gmax = fmaxf(gmax, smem[OFF_RED + i]);
    float e0 = __expf(nv0 - gmax), e1 = __expf(nv1 - gmax);
    smem[OFF_ATT + tid]       = e0;
    smem[OFF_ATT + tid + 256] = e1;
    float s = e0 + e1;
    #pragma unroll
    for (int sh = 16; sh > 0; sh >>= 1) s += __shfl_xor(s, sh, 32);
    __syncthreads();                        // done reading RED(max) & writing ATT
    if (lane == 0) smem[OFF_RED + wave] = s;
    __syncthreads();
    float gsum = 0.f;
    #pragma unroll
    for (int i = 0; i < 8; ++i) gsum += smem[OFF_RED + i];
    const float rinv = 1.0f / gsum;

    // ---------- phase 3: weighted sum of E rows (from LDS, single global pass) ----------
    {
        int h = tid & 63;
        int chunk = tid >> 6;               // 4 chunks x 128 atoms
        float acc = 0.f;
        for (int a0 = 0; a0 < 128; ++a0) {
            int row = chunk * 128 + a0;
            acc = fmaf(smem[OFF_ATT + row], sE[row * SE_STRIDE + h], acc);
        }
        smem[OFF_C + chunk * 64 + h] = acc;
    }
    __syncthreads();
    if (tid < 64) {
        float cv = (smem[OFF_C + tid] + smem[OFF_C + 64 + tid] +
                    smem[OFF_C + 128 + tid] + smem[OFF_C + 192 + tid]) * rinv;
        smem[OFF_C2 + tid] = cv;
    }
    __syncthreads();

    // ---------- phase 4: pooled = cvec @ W_lin + b_lin (softmax folds through affine) ----------
    {
        float p = blv[tid];
        #pragma unroll 8
        for (int k = 0; k < AD; ++k)
            p = fmaf(smem[OFF_C2 + k], Wl[k * NH + tid], p);
        smem[OFF_POOL + tid] = p;
    }
    __syncthreads();

    // ---------- phase 5: h = relu(pooled @ W1 + b1) ----------
    {
        int j = tid >> 5;                   // output unit 0..7, one wave each
        float acc = 0.f;
        #pragma unroll
        for (int i = 0; i < 8; ++i) {
            int k = lane + 32 * i;
            acc = fmaf(smem[OFF_POOL + k], W1[k * 8 + j], acc);
        }
        #pragma unroll
        for (int sh = 16; sh > 0; sh >>= 1) acc += __shfl_xor(acc, sh, 32);
        if (lane == 0) smem[OFF_H + j] = fmaxf(acc + b1v[j], 0.f);
    }
    __syncthreads();

    // ---------- phase 6: out = h @ W2 + b2 ----------
    if (tid < 2) {
        float o = b2v[tid];
        #pragma unroll
        for (int i = 0; i < 8; ++i)
            o = fmaf(smem[OFF_H + i], W2[i * 2 + tid], o);
        out[b * 2 + tid] = o;
    }
}

extern "C" void kernel_launch(void* const* d_in, const int* in_sizes, int n_in,
                              void* d_out, int out_size, void* d_ws, size_t ws_size,
                              hipStream_t stream) {
    const float* cpd = (const float*)d_in[0];
    // d_in[1] = target_embeddings: unused by the reference (n_layers == 0)
    const float* Wl  = (const float*)d_in[2];
    const float* blv = (const float*)d_in[3];
    const float* W1  = (const float*)d_in[4];
    const float* b1v = (const float*)d_in[5];
    const float* W2  = (const float*)d_in[6];
    const float* b2v = (const float*)d_in[7];
    float* ws  = (float*)d_ws;              // needs WS_FLOATS*4 ~= 16.6 KB of scratch
    float* out = (float*)d_out;

    it_prep_kernel<<<17, BDIM, 0, stream>>>(Wl, blv, ws);

    const size_t shmem = (size_t)LDS_FLOATS * sizeof(float);  // ~166 KB dynamic LDS
    it_fused_kernel<<<NB, BDIM, shmem, stream>>>(cpd, Wl, blv, W1, b1v, W2, b2v, ws, out);
}